// ProjectedMeshRasterizer_65592740544956
// MI455X (gfx1250) — compile-verified
//
#include <hip/hip_runtime.h>

#define IMG     128
#define KF      4
#define BLUR_R2 0.01f
#define EPSV    1e-8f
#define REC     28          // floats per face record
#define BATCH   2
#define NPIX    (BATCH*IMG*IMG)

typedef __attribute__((ext_vector_type(2))) float v2f;
typedef __attribute__((ext_vector_type(8))) float v8f;

#if defined(__AMDGCN__) && __has_builtin(__builtin_amdgcn_wmma_f32_16x16x4_f32)
#define HAVE_WMMA 1
#else
#define HAVE_WMMA 0
#endif

// ---------------- per-face record precompute ----------------
// layout: [0..2] b0 coeffs (A,B,C)*inv  [3..5] b1  [6..8] b2  [9..11] z coeffs
//         [12] area
//         [13..17] seg a0->a1: ax,ay,abx,aby,inv(|ab|^2+eps)
//         [18..22] seg a1->a2   [23..27] seg a2->a0
__device__ __forceinline__ void seg_rec(float* r, float ax, float ay, float bx, float by) {
    float abx = bx - ax, aby = by - ay;
    r[0] = ax; r[1] = ay; r[2] = abx; r[3] = aby;
    r[4] = 1.0f / (abx*abx + aby*aby + EPSV);
}

__global__ void prep_faces(const float* __restrict__ verts, const int* __restrict__ faces,
                           float* __restrict__ recs, int V, int F) {
    int id = blockIdx.x * blockDim.x + threadIdx.x;
    if (id >= BATCH * F) return;
    int b = id / F, f = id - b * F;
    const float* vb = verts + (size_t)b * V * 3;
    int i0 = faces[f*3+0], i1 = faces[f*3+1], i2 = faces[f*3+2];
    float x0 = vb[i0*3+0], y0 = vb[i0*3+1], z0 = vb[i0*3+2];
    float x1 = vb[i1*3+0], y1 = vb[i1*3+1], z1 = vb[i1*3+2];
    float x2 = vb[i2*3+0], y2 = vb[i2*3+1], z2 = vb[i2*3+2];

    // w0 = edge(a1,a2,p) = px*(y2-y1) - py*(x2-x1) + (y1*(x2-x1) - x1*(y2-y1))
    float A0 = y2 - y1, B0 = -(x2 - x1), C0 = y1*(x2-x1) - x1*(y2-y1);
    float A1 = y0 - y2, B1 = -(x0 - x2), C1 = y2*(x0-x2) - x2*(y0-y2);
    float A2 = y1 - y0, B2 = -(x1 - x0), C2 = y0*(x1-x0) - x0*(y1-y0);
    float area = (x2-x0)*(y1-y0) - (y2-y0)*(x1-x0);
    float denom = (fabsf(area) > EPSV) ? area : ((area >= 0.0f) ? EPSV : -EPSV);
    float inv = 1.0f / denom;

    float* r = recs + (size_t)id * REC;
    r[0] = A0*inv; r[1] = B0*inv; r[2] = C0*inv;
    r[3] = A1*inv; r[4] = B1*inv; r[5] = C1*inv;
    r[6] = A2*inv; r[7] = B2*inv; r[8] = C2*inv;
    r[9]  = z0*r[0] + z1*r[3] + z2*r[6];
    r[10] = z0*r[1] + z1*r[4] + z2*r[7];
    r[11] = z0*r[2] + z1*r[5] + z2*r[8];
    r[12] = area;
    seg_rec(r+13, x0, y0, x1, y1);
    seg_rec(r+18, x1, y1, x2, y2);
    seg_rec(r+23, x2, y2, x0, y0);
}

// ---------------- raster kernel ----------------
struct Ent { float z; int f; float b0, b1, b2, d; };

__device__ __forceinline__ bool lessE(const Ent& a, const Ent& b) {
    return (a.z < b.z) || (a.z == b.z && a.f < b.f);   // JAX top_k tie-break: lower index first
}
__device__ __forceinline__ void cex(Ent& a, Ent& b) {
    if (lessE(b, a)) { Ent t = a; a = b; b = t; }
}

__device__ __forceinline__ float segd2(float px, float py, const float* s) {
    float pax = px - s[0], pay = py - s[1];
    float abx = s[2], aby = s[3];
    float tt = fmaf(pax, abx, pay * aby) * s[4];
    tt = fminf(fmaxf(tt, 0.0f), 1.0f);
    float dx = fmaf(-tt, abx, pax);
    float dy = fmaf(-tt, aby, pay);
    return fmaf(dx, dx, dy * dy);
}

__global__ void raster(const float* __restrict__ recs, float* __restrict__ out, int F) {
    extern __shared__ float sface[];               // F * REC floats (89.6 KB @ F=800; WGP has 320 KB)
    int b = blockIdx.y;
    int total = F * REC;
    for (int i = threadIdx.x; i < total; i += blockDim.x)
        sface[i] = recs[(size_t)b * total + i];
    __syncthreads();

    int lane = threadIdx.x & 31;                   // wave32
    int wave = threadIdx.x >> 5;
    bool hi  = lane >= 16;
    int pix  = blockIdx.x * 128 + wave * 16 + (lane & 15);  // 8 waves * 16 pixels
    int ix = pix & (IMG - 1);
    int iy = pix >> 7;
    float px = 1.0f - 2.0f * ((float)ix + 0.5f) / (float)IMG;
    float py = 1.0f - 2.0f * ((float)iy + 0.5f) / (float)IMG;

    Ent t[4];
    #pragma unroll
    for (int i = 0; i < 4; ++i) t[i] = Ent{__builtin_inff(), -1, -1.f, -1.f, -1.f, -1.f};

#if HAVE_WMMA
    // B matrix 4x16 (KxN): K0=px[N], K1=py[N], K2=1, K3=0.
    // VGPR0: lanes 0-15 row K0, lanes 16-31 row K1; VGPR1: K2 / K3.
    v2f bm; bm.x = hi ? py : px; bm.y = hi ? 0.0f : 1.0f;
    v8f zacc = {};
#endif

    int nsteps = F >> 4;
    for (int s = 0; s < nsteps; ++s) {
        const float* fr = &sface[(s * 16 + (lane & 15)) * REC];
#if HAVE_WMMA
        // A matrix 16x4 (MxK): lanes 0-15 carry K0,K1 of face M=lane; lanes 16-31 carry K2,K3.
        v2f a0v, a1v, a2v, azv;
        a0v.x = hi ? fr[2]  : fr[0];  a0v.y = hi ? 0.0f : fr[1];
        a1v.x = hi ? fr[5]  : fr[3];  a1v.y = hi ? 0.0f : fr[4];
        a2v.x = hi ? fr[8]  : fr[6];  a2v.y = hi ? 0.0f : fr[7];
        azv.x = hi ? fr[11] : fr[9];  azv.y = hi ? 0.0f : fr[10];
        // D[M=face, N=pixel] = c0*px + c1*py + c2 : 4 linear forms per 16x16 tile.
        v8f cb0 = __builtin_amdgcn_wmma_f32_16x16x4_f32(false, a0v, false, bm, (short)0, zacc, false, false);
        v8f cb1 = __builtin_amdgcn_wmma_f32_16x16x4_f32(false, a1v, false, bm, (short)0, zacc, false, false);
        v8f cb2 = __builtin_amdgcn_wmma_f32_16x16x4_f32(false, a2v, false, bm, (short)0, zacc, false, false);
        v8f cz  = __builtin_amdgcn_wmma_f32_16x16x4_f32(false, azv, false, bm, (short)0, zacc, false, false);
#endif
        #pragma unroll
        for (int r = 0; r < 8; ++r) {
            int fid = s * 16 + (hi ? 8 : 0) + r;   // D rows: lanes<16 -> M=0..7, lanes>=16 -> M=8..15
            const float* rc = &sface[fid * REC];
#if HAVE_WMMA
            float b0 = cb0[r], b1 = cb1[r], b2 = cb2[r], zp = cz[r];
#else
            float b0 = fmaf(px, rc[0], fmaf(py, rc[1],  rc[2]));
            float b1 = fmaf(px, rc[3], fmaf(py, rc[4],  rc[5]));
            float b2 = fmaf(px, rc[6], fmaf(py, rc[7],  rc[8]));
            float zp = fmaf(px, rc[9], fmaf(py, rc[10], rc[11]));
#endif
            bool inside = (b0 >= 0.0f) && (b1 >= 0.0f) && (b2 >= 0.0f);
            float dmin = fminf(segd2(px, py, rc+13),
                         fminf(segd2(px, py, rc+18), segd2(px, py, rc+23)));
            float dsg = inside ? -dmin : dmin;
            bool valid = (inside || (dmin < BLUR_R2)) && (zp > EPSV) && (fabsf(rc[12]) > EPSV);
            if (valid) {
                Ent c{zp, fid, b0, b1, b2, dsg};
                if (lessE(c, t[3])) { t[3] = c; cex(t[2], t[3]); cex(t[1], t[2]); cex(t[0], t[1]); }
            }
        }
    }

    // scalar tail for F not multiple of 16 (empty at F=800); low half only to avoid double count
    for (int fid = nsteps * 16; fid < F; ++fid) {
        if (hi) break;
        const float* rc = &sface[fid * REC];
        float b0 = fmaf(px, rc[0], fmaf(py, rc[1],  rc[2]));
        float b1 = fmaf(px, rc[3], fmaf(py, rc[4],  rc[5]));
        float b2 = fmaf(px, rc[6], fmaf(py, rc[7],  rc[8]));
        float zp = fmaf(px, rc[9], fmaf(py, rc[10], rc[11]));
        bool inside = (b0 >= 0.0f) && (b1 >= 0.0f) && (b2 >= 0.0f);
        float dmin = fminf(segd2(px, py, rc+13),
                     fminf(segd2(px, py, rc+18), segd2(px, py, rc+23)));
        float dsg = inside ? -dmin : dmin;
        bool valid = (inside || (dmin < BLUR_R2)) && (zp > EPSV) && (fabsf(rc[12]) > EPSV);
        if (valid) {
            Ent c{zp, fid, b0, b1, b2, dsg};
            if (lessE(c, t[3])) { t[3] = c; cex(t[2], t[3]); cex(t[1], t[2]); cex(t[0], t[1]); }
        }
    }

    // merge sorted top-4 of lane L with lane L+16 (Batcher 4+4 odd-even merge, 9 CEs)
    Ent c8[8];
    #pragma unroll
    for (int i = 0; i < 4; ++i) c8[i] = t[i];
    #pragma unroll
    for (int i = 0; i < 4; ++i) {
        Ent e;
        e.z  = __shfl_xor(t[i].z,  16, 32);
        e.f  = __shfl_xor(t[i].f,  16, 32);
        e.b0 = __shfl_xor(t[i].b0, 16, 32);
        e.b1 = __shfl_xor(t[i].b1, 16, 32);
        e.b2 = __shfl_xor(t[i].b2, 16, 32);
        e.d  = __shfl_xor(t[i].d,  16, 32);
        c8[4 + i] = e;
    }
    cex(c8[0], c8[4]); cex(c8[1], c8[5]); cex(c8[2], c8[6]); cex(c8[3], c8[7]);
    cex(c8[2], c8[4]); cex(c8[3], c8[5]);
    cex(c8[1], c8[2]); cex(c8[3], c8[4]); cex(c8[5], c8[6]);

    if (!hi) {
        size_t pg = (size_t)b * IMG * IMG + (size_t)pix;
        #pragma unroll
        for (int k = 0; k < 4; ++k) {
            Ent e = c8[k];
            bool ok = e.f >= 0;
            out[                 pg * 4  + k]         = ok ? (float)e.f : -1.0f;  // pix_to_face
            out[(size_t)NPIX*4 + pg * 4  + k]         = ok ? e.z        : -1.0f;  // zbuf
            out[(size_t)NPIX*8 + pg * 12 + k*3 + 0]   = ok ? e.b0       : -1.0f;  // bary
            out[(size_t)NPIX*8 + pg * 12 + k*3 + 1]   = ok ? e.b1       : -1.0f;
            out[(size_t)NPIX*8 + pg * 12 + k*3 + 2]   = ok ? e.b2       : -1.0f;
            out[(size_t)NPIX*20+ pg * 4  + k]         = ok ? e.d        : -1.0f;  // dists
        }
    }
}

extern "C" void kernel_launch(void* const* d_in, const int* in_sizes, int n_in,
                              void* d_out, int out_size, void* d_ws, size_t ws_size,
                              hipStream_t stream) {
    const float* verts = (const float*)d_in[0];
    const int*   faces = (const int*)d_in[1];
    int V = in_sizes[0] / (BATCH * 3);
    int F = in_sizes[1] / 3;

    float* recs = (float*)d_ws;                      // BATCH*F*REC floats (~175 KB)
    int nrec = BATCH * F;
    prep_faces<<<(nrec + 255) / 256, 256, 0, stream>>>(verts, faces, recs, V, F);

    dim3 grid(IMG * IMG / 128, BATCH);               // 8 waves/block, 16 pixels/wave
    size_t smem = (size_t)F * REC * sizeof(float);   // 89.6 KB dyn-LDS; WGP supports 320 KB
    raster<<<grid, 256, smem, stream>>>(recs, (float*)d_out, F);
}